// YOLOv2Loss_36455682408970
// MI455X (gfx1250) — compile-verified
//
#include <hip/hip_runtime.h>
#include <hip/hip_bf16.h>
#include <math.h>
#include <stdint.h>

typedef __attribute__((ext_vector_type(2))) float        v2f;
typedef __attribute__((ext_vector_type(8))) float        v8f;
typedef __attribute__((ext_vector_type(4))) unsigned int v4u;
typedef __attribute__((ext_vector_type(8))) int          v8i;
typedef __attribute__((ext_vector_type(4))) int          v4i;

#define BSZ   64
#define NA    5
#define NCLS  80
#define NO    85
#define NXG   19
#define NYG   19
#define HW    (NXG*NYG)          // 361
#define NT    512
#define NCELL (BSZ*NA*HW)        // 115520
#define NEGV  (-1e30f)
#define IGN   0.5f
#define STAGE_DW (NT*4 + NT)     // 2560 dwords: gt (512 float4) + img (512 int)

// Faithful replica of reference _iou_cxcywh, including the buggy
// area = (w - cx) * (h - cy) term.
__device__ __forceinline__ float iou_ref(float ax, float ay, float aw, float ah,
                                         float bx, float by, float bw, float bh) {
    float tlx = fmaxf(ax - aw * 0.5f, bx - bw * 0.5f);
    float tly = fmaxf(ay - ah * 0.5f, by - bh * 0.5f);
    float brx = fminf(ax + aw * 0.5f, bx + bw * 0.5f);
    float bry = fminf(ay + ah * 0.5f, by + bh * 0.5f);
    float area_a = (aw - ax) * (ah - ay);
    float area_b = (bw - bx) * (bh - by);
    float ai = ((tlx < brx) && (tly < bry)) ? (brx - tlx) * (bry - tly) : 0.0f;
    return ai / (area_a + area_b - ai);
}

__device__ __forceinline__ float sigmoidf_(float x) { return 1.0f / (1.0f + expf(-x)); }

// ---------------------------------------------------------------- init
__global__ void k_init(int* __restrict__ winner, int* __restrict__ posex) {
    int i = blockIdx.x * blockDim.x + threadIdx.x;
    if (i < NCELL) winner[i] = -1;          // empty segment_max -> unassigned
    if (i < BSZ)   posex[i]  = 0;
}

// ------------------------------------------------------- target preprocessing
__global__ void k_prep(const float* __restrict__ targets, const float* __restrict__ anchors,
                       float4* __restrict__ gt, int* __restrict__ img, int* __restrict__ cls,
                       int* __restrict__ besta, int* __restrict__ flatidx,
                       float4* __restrict__ deltas, int* __restrict__ winner) {
    int t = threadIdx.x;
    if (t >= NT) return;
    const float* tr = targets + t * 6;
    int   im = (int)tr[0];
    int   cl = (int)tr[1];
    float gx = tr[2] * (float)NXG, gy = tr[3] * (float)NYG;
    float gw = tr[4] * (float)NXG, gh = tr[5] * (float)NYG;
    float cxf = floorf(gx), cyf = floorf(gy);
    int cell = (int)cyf * NXG + (int)cxf;

    // best_a = argmax over anchors of (int)iou  (truncated, first-max wins)
    int best = 0, bestv = -2147483647 - 1;
    for (int a = 0; a < NA; ++a) {
        float io = iou_ref(cxf, cyf, anchors[2*a], anchors[2*a+1], gx, gy, gw, gh);
        if (io != io) io = 0.0f;                          // NaN -> 0 like jnp astype
        io = fminf(fmaxf(io, -2.0e9f), 2.0e9f);           // avoid UB on cast
        int vi = (int)io;
        if (vi > bestv) { bestv = vi; best = a; }
    }
    float aw = anchors[2*best], ah = anchors[2*best+1];

    gt[t]      = make_float4(gx, gy, gw, gh);
    img[t]     = im;
    cls[t]     = cl;
    besta[t]   = best;
    deltas[t]  = make_float4(gx - cxf, gy - cyf, gw / aw, gh / ah);
    int fl     = (im * NA + best) * HW + cell;
    flatidx[t] = fl;
    atomicMax(&winner[fl], t);                            // segment_max(arange(nt), flat)
}

// ---------------------------------------------- max IoU over targets per cell
// Targets table (gt + img, 10 KB, contiguous in workspace) is DMA'd into LDS
// by the Tensor Data Mover: one tensor_load_to_lds issued by wave 0, then
// s_wait_tensorcnt 0 + workgroup barrier.
__global__ void __launch_bounds__(256)
k_maxiou(const float* __restrict__ p, const float* __restrict__ anchors,
         const float4* __restrict__ gt, const int* __restrict__ img,
         float* __restrict__ maxiou, int* __restrict__ posex) {
    __shared__ float s_buf[STAGE_DW];                  // [0,2048): gt, [2048,2560): img
    float4* s_gt  = (float4*)s_buf;
    int*    s_img = (int*)(s_buf + NT * 4);
    int tid = threadIdx.x;

#if __has_builtin(__builtin_amdgcn_tensor_load_to_lds)
    if (tid < 32) {
        uint32_t lds_addr = (uint32_t)(uintptr_t)(void*)&s_buf[0];   // low 32b = LDS offset
        uint64_t ga = (uint64_t)(uintptr_t)(const void*)gt;          // gt||img contiguous
        const uint32_t n = STAGE_DW;                                 // elements (4B each)
        v4u g0;
        g0.x = 1u;                                                   // count=1, user mode
        g0.y = lds_addr;                                             // D#.lds_addr
        g0.z = (uint32_t)(ga & 0xffffffffu);                         // global_addr[31:0]
        g0.w = (uint32_t)((ga >> 32) & 0x01ffffffu) | (2u << 30);    // ga[56:32] | type=2
        v8i g1;
        g1[0] = (int)(2u << 16);                 // workgroup_mask=0, data_size=4B
        g1[1] = (int)((n & 0xffffu) << 16);      // tensor_dim0[15:0]
        g1[2] = (int)(((n >> 16) & 0xffffu) | (1u << 16)); // tensor_dim0[31:16], tensor_dim1=1
        g1[3] = (int)((n & 0xffffu) << 16);      // tensor_dim1[31:16]=0, tile_dim0=n
        g1[4] = 1;                               // tile_dim1=1, tile_dim2=0
        g1[5] = (int)n;                          // tensor_dim0_stride[31:0]
        g1[6] = 0;                               // stride0[47:32]=0, stride1[15:0]=0
        g1[7] = 0;
        v4i g2 = {0, 0, 0, 0};                   // <=2D tensor: groups 2/3 zero
        v4i g3 = {0, 0, 0, 0};
#if defined(__clang_major__) && (__clang_major__ >= 23)
        v8i g4 = {0, 0, 0, 0, 0, 0, 0, 0};
        __builtin_amdgcn_tensor_load_to_lds(g0, g1, g2, g3, g4, 0);
#else
        __builtin_amdgcn_tensor_load_to_lds(g0, g1, g2, g3, 0);
#endif
        __builtin_amdgcn_s_wait_tensorcnt(0);    // issuing wave drains its TENSORcnt
    }
#else
    for (int j = tid; j < NT; j += blockDim.x) { s_gt[j] = gt[j]; s_img[j] = img[j]; }
#endif
    __syncthreads();

    int b = blockIdx.y;
    int j = blockIdx.x * blockDim.x + tid;
    if (j >= NA * HW) return;
    int a = j / HW, cell = j % HW;
    int i = (b * NA + a) * HW + cell;
    size_t base = ((size_t)(b * NA + a) * NO) * HW + (size_t)cell;
    __builtin_prefetch(p + base, 0, 0);                   // global_prefetch_b8

    float x0 = p[base + 0 * HW], x1 = p[base + 1 * HW];
    float x2 = p[base + 2 * HW], x3 = p[base + 3 * HW];
    float px = sigmoidf_(x0) + (float)(cell % NXG);
    float py = sigmoidf_(x1) + (float)(cell / NXG);
    float pw = expf(x2) * anchors[2*a];
    float ph = expf(x3) * anchors[2*a+1];
    float ax0 = px - pw * 0.5f, ax1 = px + pw * 0.5f;
    float ay0 = py - ph * 0.5f, ay1 = py + ph * 0.5f;
    float area_a = (pw - px) * (ph - py);

    float best = NEGV;
    #pragma unroll 4
    for (int t = 0; t < NT; ++t) {
        float4 g = s_gt[t];
        float tlx = fmaxf(ax0, g.x - g.z * 0.5f);
        float tly = fmaxf(ay0, g.y - g.w * 0.5f);
        float brx = fminf(ax1, g.x + g.z * 0.5f);
        float bry = fminf(ay1, g.y + g.w * 0.5f);
        float area_b = (g.z - g.x) * (g.w - g.y);
        float ai  = ((tlx < brx) && (tly < bry)) ? (brx - tlx) * (bry - tly) : 0.0f;
        float iou = ai / (area_a + area_b - ai);
        best = fmaxf(best, (s_img[t] == b) ? iou : NEGV);
    }
    maxiou[i] = best;
    if (best > IGN) atomicOr(&posex[b], 1);               // pos_exists (strict >)
}

// ----------------------------------------- per-target scale_t / iou target
__global__ void k_tgt2(const float* __restrict__ p, const float* __restrict__ anchors,
                       const int* __restrict__ img, const int* __restrict__ besta,
                       const int* __restrict__ flatidx, const float* __restrict__ maxiou,
                       float* __restrict__ scale_t, float* __restrict__ ioutgt) {
    int t = blockIdx.x * blockDim.x + threadIdx.x;
    if (t >= NT) return;
    int fl = flatidx[t], b = img[t], a = besta[t];
    int cell = fl % HW;
    size_t base = ((size_t)(b * NA + a) * NO) * HW + (size_t)cell;
    float pw = expf(p[base + 2 * HW]) * anchors[2*a];
    float ph = expf(p[base + 3 * HW]) * anchors[2*a+1];
    scale_t[t] = 2.0f - (pw / (float)NXG) * (ph / (float)NYG);
    ioutgt[t]  = maxiou[fl];
}

// ---------------------------------------------------------- main loss pass
__global__ void __launch_bounds__(256)
k_loss(const float* __restrict__ p, const int* __restrict__ winner,
       const float* __restrict__ maxiou, const int* __restrict__ posex,
       const float4* __restrict__ deltas, const float* __restrict__ scale_t,
       const float* __restrict__ ioutgt, const int* __restrict__ cls,
       float* __restrict__ partials) {
    int tid = threadIdx.x;
    int i = blockIdx.x * 256 + tid;
    float q[6] = {0.f, 0.f, 0.f, 0.f, 0.f, 0.f};  // Sbox, nbox, Sobj, Snoobj, Cnoobj, Sce
    if (i < NCELL) {
        int b = i / (NA * HW);
        int rem = i % (NA * HW);
        int a = rem / HW, cell = rem % HW;
        size_t base = ((size_t)(b * NA + a) * NO) * HW + (size_t)cell;
        float conf = sigmoidf_(p[base + 4 * HW]);
        int  w = winner[i];
        bool assigned = (w >= 0);
        int  wt = assigned ? w : 0;
        bool ign = (maxiou[i] >= IGN) && (posex[b] != 0);
        float itgt = assigned ? ioutgt[wt] : 0.0f;
        float sq = conf - itgt; sq *= sq;
        if (assigned)       { q[1] = 1.0f; q[2] = sq; }   // iou_mask == 2
        else if (!ign)      { q[3] = sq;  q[4] = 1.0f; }  // iou_mask == 1
        if (assigned) {
            float4 bt = deltas[wt];
            float s = sqrtf(scale_t[wt]);
            float bp0 = sigmoidf_(p[base + 0 * HW]) * s - bt.x;
            float bp1 = sigmoidf_(p[base + 1 * HW]) * s - bt.y;
            float bp2 = expf(p[base + 2 * HW])      * s - bt.z;
            float bp3 = expf(p[base + 3 * HW])      * s - bt.w;
            q[0] = bp0*bp0 + bp1*bp1 + bp2*bp2 + bp3*bp3;
            int ct = cls[wt];
            float mx = -3.4e38f;
            for (int o = 0; o < NCLS; ++o) mx = fmaxf(mx, p[base + (size_t)(5 + o) * HW]);
            float se = 0.0f, xt = 0.0f;
            for (int o = 0; o < NCLS; ++o) {
                float v = p[base + (size_t)(5 + o) * HW];
                se += expf(v - mx);
                if (o == ct) xt = v;
            }
            q[5] = (mx + logf(se)) - xt;                  // -log_softmax[ct]
        }
    }
    // wave32 tree reduction
    #pragma unroll
    for (int qi = 0; qi < 6; ++qi)
        #pragma unroll
        for (int off = 16; off >= 1; off >>= 1)
            q[qi] += __shfl_xor(q[qi], off, 32);

    __shared__ float part[8][6];
    int lane = tid & 31, wid = tid >> 5;
    if (lane == 0)
        for (int qi = 0; qi < 6; ++qi) part[wid][qi] = q[qi];
    __syncthreads();

    // WMMA ones-matmul reduction of the 8x6 per-wave partials.
    // A(16x4) rows m=quantity carry part[wave k][m]; B = ones(4x16)
    // => D[m,n] = sum_k A[m][k]. Two chained WMMAs cover waves 0..3 and 4..7.
    // Every wave executes it (EXEC all-ones requirement); lane 0's D VGPRs
    // 0..5 hold the six block sums (D layout: VGPR v, lanes 0-15 -> M=v, N=lane).
    int m = lane & 15;
    int k0 = (lane >= 16) ? 2 : 0;
    v2f A1, A2, B;
    A1.x = (m < 6) ? part[k0 + 0][m] : 0.0f;
    A1.y = (m < 6) ? part[k0 + 1][m] : 0.0f;
    A2.x = (m < 6) ? part[4 + k0 + 0][m] : 0.0f;
    A2.y = (m < 6) ? part[4 + k0 + 1][m] : 0.0f;
    B.x = 1.0f; B.y = 1.0f;
    v8f C = {};
    C = __builtin_amdgcn_wmma_f32_16x16x4_f32(false, A1, false, B, (short)0, C, false, false);
    C = __builtin_amdgcn_wmma_f32_16x16x4_f32(false, A2, false, B, (short)0, C, false, false);
    if (tid == 0)
        for (int qi = 0; qi < 6; ++qi) partials[blockIdx.x * 6 + qi] = C[qi];
}

// --------------------------------------------------- final deterministic sum
__global__ void k_final(const float* __restrict__ partials, int nblocks, float* __restrict__ out) {
    __builtin_amdgcn_s_wait_tensorcnt(0);
    int lane = threadIdx.x;                                // 32 threads, one wave
    float acc[6] = {0.f, 0.f, 0.f, 0.f, 0.f, 0.f};
    for (int bb = lane; bb < nblocks; bb += 32)
        for (int qi = 0; qi < 6; ++qi) acc[qi] += partials[bb * 6 + qi];
    #pragma unroll
    for (int qi = 0; qi < 6; ++qi)
        #pragma unroll
        for (int off = 16; off >= 1; off >>= 1)
            acc[qi] += __shfl_xor(acc[qi], off, 32);
    if (lane == 0) {
        float nbox       = acc[1];
        float box_loss   = acc[0] / (4.0f * nbox);
        float obj_loss   = acc[2] / nbox;
        float noobj_loss = acc[3] / acc[4];
        float class_loss = acc[5] / nbox;
        float lbox = 5.0f * box_loss;
        float lobj = 5.0f * obj_loss + 1.0f * noobj_loss;
        float lcls = 1.0f * class_loss;
        out[0] = lbox + lobj + lcls;
        out[1] = lbox; out[2] = lobj; out[3] = lcls;
    }
}

extern "C" void kernel_launch(void* const* d_in, const int* in_sizes, int n_in,
                              void* d_out, int out_size, void* d_ws, size_t ws_size,
                              hipStream_t stream) {
    (void)in_sizes; (void)n_in; (void)out_size; (void)ws_size;
    const float* p       = (const float*)d_in[0];
    const float* targets = (const float*)d_in[1];
    const float* anchors = (const float*)d_in[2];
    float* out = (float*)d_out;

    char* ws = (char*)d_ws;
    float4* gt      = (float4*)(ws + 0);                   //  8192 B  } contiguous 10240 B
    int*    img     = (int*)   (ws + 8192);                //  2048 B  } staged by one TDM op
    int*    cls     = (int*)   (ws + 10240);               //  2048 B
    int*    besta   = (int*)   (ws + 12288);               //  2048 B
    int*    flatidx = (int*)   (ws + 14336);               //  2048 B
    float4* deltas  = (float4*)(ws + 16384);               //  8192 B
    float*  scale_t = (float*) (ws + 24576);               //  2048 B
    float*  ioutgt  = (float*) (ws + 26624);               //  2048 B
    int*    posex   = (int*)   (ws + 28672);               //   256 B
    float*  partials= (float*) (ws + 28928);               // 10848 B
    int*    winner  = (int*)   (ws + 40960);               // 462080 B
    float*  maxiou  = (float*) (ws + 40960 + (size_t)NCELL * 4);

    const int LB = (NCELL + 255) / 256;                    // 452 loss blocks

    k_init  <<<LB, 256, 0, stream>>>(winner, posex);
    k_prep  <<<1, NT, 0, stream>>>(targets, anchors, gt, img, cls, besta, flatidx, deltas, winner);
    k_maxiou<<<dim3((NA * HW + 255) / 256, BSZ), 256, 0, stream>>>(p, anchors, gt, img, maxiou, posex);
    k_tgt2  <<<(NT + 255) / 256, 256, 0, stream>>>(p, anchors, img, besta, flatidx, maxiou, scale_t, ioutgt);
    k_loss  <<<LB, 256, 0, stream>>>(p, winner, maxiou, posex, deltas, scale_t, ioutgt, cls, partials);
    k_final <<<1, 32, 0, stream>>>(partials, LB, out);
}